// GatedDeltaRuleGQAModel_65017214926849
// MI455X (gfx1250) — compile-verified
//
#include <hip/hip_runtime.h>
#include <hip/hip_bf16.h>
#include <math.h>

// Gated delta rule, exact chunked WY-form, fp32 WMMA (gfx1250).
// Grid: B*HV*(DV/64) = 128 workgroups of 128 threads (4 waves, wave32).
// Each wave owns a 16-row band of the state S[64-band][128].
// v2: software-pipelined global->reg->LDS staging, reciprocal decay table.

typedef __attribute__((ext_vector_type(2))) float v2f;
typedef __attribute__((ext_vector_type(8))) float v8f;

#define BB   2
#define TT   4096
#define HKC  16
#define HVC  32
#define DKC  128
#define DVC  128
#define CHK  16     // chunk (tokens)
#define DVB  64     // DV rows per workgroup
#define NTHR 128
#define NCH  (TT / CHK)
#define KSTR 132    // padded stride for [*][128] tiles (bank-conflict-free A reads)
#define VSTR 68
#define TSTR 17

__device__ __forceinline__ v8f wmma4(v2f a, v2f b, v8f c) {
  // V_WMMA_F32_16X16X4_F32 : D = A(16x4) * B(4x16) + C(16x16), all f32
  return __builtin_amdgcn_wmma_f32_16x16x4_f32(false, a, false, b, (short)0, c,
                                               false, false);
}

__global__ __launch_bounds__(NTHR) void gdn_chunked_kernel(
    const float* __restrict__ q, const float* __restrict__ k,
    const float* __restrict__ v, const float* __restrict__ g,
    const float* __restrict__ beta, const float* __restrict__ state,
    float* __restrict__ out)
{
  __shared__ float sS[DVB * KSTR];     // state band [64][128]
  __shared__ float sK[CHK * KSTR];     // K chunk   [16][128]
  __shared__ float sQ[CHK * KSTR];     // Q chunk   [16][128]
  __shared__ float sV[CHK * VSTR];     // V chunk   [16][64]
  __shared__ float sG[CHK * TSTR];     // K K^T
  __shared__ float sHq[CHK * TSTR];    // Q K^T
  __shared__ float sT[CHK * TSTR];     // (I+W)^-1
  __shared__ float sM[CHK * TSTR];     // causal decay-weighted QK^T
  __shared__ float sScr[4][CHK * TSTR];// per-wave U/Delta scratch
  __shared__ float sGr[CHK];           // raw g
  __shared__ float sBet[CHK];          // beta
  __shared__ float sCp[CHK];           // cp_i = exp(cumsum g) inclusive
  __shared__ float sRcp[CHK];          // 1 / cp_i
  __shared__ float sDs[CHK];           // cp_15 / cp_i
  __shared__ float sCpT;               // cp_15

  const int tid  = threadIdx.x;
  const int wave = tid >> 5;
  const int lane = tid & 31;
  const int half = lane >> 4;
  const int lid  = lane & 15;
  const int koff = 2 * half;

  const int wg   = blockIdx.x;
  const int b    = wg / (HVC * 2);
  const int rem  = wg % (HVC * 2);
  const int h    = rem >> 1;
  const int dblk = rem & 1;
  const int hk   = h >> 1;             // head_repeat = HV/HK = 2
  const int rowOff = dblk * DVB;

  const size_t qkTok = (size_t)HKC * DKC;  // 2048 floats / token
  const size_t vTok  = (size_t)HVC * DVC;  // 4096 floats / token
  const float* qB  = q + ((size_t)b * TT * HKC + hk) * DKC;
  const float* kB  = k + ((size_t)b * TT * HKC + hk) * DKC;
  const float* vB  = v + ((size_t)b * TT * HVC + h) * DVC + rowOff;
  const float* gB  = g + (size_t)b * TT * HVC + h;
  const float* bB  = beta + (size_t)b * TT * HVC + h;
  float*       oB  = out + ((size_t)b * TT * HVC + h) * DVC + rowOff;
  const float* stB = state + (((size_t)b * HVC + h) * DVC + rowOff) * DKC;

  const int itok = tid >> 3;           // token this thread stages
  const int part = tid & 7;

  // staging registers for the next chunk (software pipeline)
  float4 rK[4], rQ[4], rV0, rV1;
  float  rGg = 0.f, rBt = 0.f;

  auto loadRegs = [&](int t0) {
    const float* ks = kB + (size_t)(t0 + itok) * qkTok + part * 16;
    const float* qs = qB + (size_t)(t0 + itok) * qkTok + part * 16;
#pragma unroll
    for (int j = 0; j < 4; ++j) {
      rK[j] = *(const float4*)(ks + j * 4);
      rQ[j] = *(const float4*)(qs + j * 4);
    }
    const float* vs = vB + (size_t)(t0 + itok) * vTok + part * 8;
    rV0 = *(const float4*)(vs + 0);
    rV1 = *(const float4*)(vs + 4);
    if (tid < CHK)          rGg = gB[(size_t)(t0 + tid) * HVC];
    else if (tid < 2 * CHK) rBt = bB[(size_t)(t0 + tid - CHK) * HVC];
  };

  auto storeRegs = [&]() {
    float* kd = &sK[itok * KSTR + part * 16];
    float* qd = &sQ[itok * KSTR + part * 16];
#pragma unroll
    for (int j = 0; j < 4; ++j) {
      *(float4*)(kd + j * 4) = rK[j];
      *(float4*)(qd + j * 4) = rQ[j];
    }
    float* vd = &sV[itok * VSTR + part * 8];
    *(float4*)(vd + 0) = rV0;
    *(float4*)(vd + 4) = rV1;
    if (tid < CHK)          sGr[tid]        = rGg;
    else if (tid < 2 * CHK) sBet[tid - CHK] = rBt;
  };

  // ---- prologue: issue chunk-0 loads, stage initial state -> LDS ----
  loadRegs(0);
  {
    const int row = tid >> 1;
    const int hs  = (tid & 1) * 64;
    const float* src = stB + (size_t)row * DKC + hs;
    float* dst = &sS[row * KSTR + hs];
#pragma unroll
    for (int j = 0; j < 16; ++j)
      *(float4*)(dst + j * 4) = *(const float4*)(src + j * 4);
  }

  for (int c = 0; c < NCH; ++c) {
    const int t0 = c * CHK;
    __syncthreads();  // (1) previous chunk's LDS fully consumed

    storeRegs();                       // chunk c : regs -> LDS
    if (c + 1 < NCH) loadRegs((c + 1) * CHK);   // issue chunk c+1 loads

    __syncthreads();  // (2)

    // ---- phase A: G=KK^T (w0), H=QK^T (w1), decay scalars (w2), prefetch (w3)
    if (wave == 0) {
      v8f acc = {};
#pragma unroll 8
      for (int kk = 0; kk < DKC; kk += 4) {
        v2f a; a.x = sK[lid * KSTR + kk + koff]; a.y = sK[lid * KSTR + kk + koff + 1];
        acc = wmma4(a, a, acc);   // B operand == A operand lane layout for K K^T
      }
#pragma unroll
      for (int m = 0; m < 8; ++m) sG[(m + 8 * half) * TSTR + lid] = acc[m];
    } else if (wave == 1) {
      v8f acc = {};
#pragma unroll 8
      for (int kk = 0; kk < DKC; kk += 4) {
        v2f a;  a.x  = sQ[lid * KSTR + kk + koff]; a.y  = sQ[lid * KSTR + kk + koff + 1];
        v2f bb; bb.x = sK[lid * KSTR + kk + koff]; bb.y = sK[lid * KSTR + kk + koff + 1];
        acc = wmma4(a, bb, acc);
      }
#pragma unroll
      for (int m = 0; m < 8; ++m) sHq[(m + 8 * half) * TSTR + lid] = acc[m];
    } else if (wave == 2) {
      if (lane < CHK) {
        float si = 0.f, st = 0.f;
#pragma unroll
        for (int j = 0; j < CHK; ++j) {
          float gg = sGr[j];
          st += gg;
          si += (j <= lane) ? gg : 0.f;
        }
        sCp[lane]  = expf(si);
        sRcp[lane] = expf(-si);
        sDs[lane]  = expf(st - si);
        if (lane == CHK - 1) sCpT = expf(st);
      }
    } else {
      // warm L2 two chunks ahead (loads for c+1 are already in flight)
      if (t0 + 2 * CHK < TT && lane < CHK) {
        const size_t tn = (size_t)(t0 + 2 * CHK + lane);
        __builtin_prefetch(kB + tn * qkTok, 0, 1);
        __builtin_prefetch(kB + tn * qkTok + 64, 0, 1);
        __builtin_prefetch(qB + tn * qkTok, 0, 1);
        __builtin_prefetch(qB + tn * qkTok + 64, 0, 1);
        __builtin_prefetch(vB + tn * vTok, 0, 1);
      }
    }
    __syncthreads();  // (3)

    // ---- phase B: T=(I+W)^-1 (w0), M matrix (w1); all waves: C1=S0K^T, P=S0Q^T
    if (wave == 0 && lane < CHK) {
      // forward substitution, column `lane` of T; W[i][m]=beta_i*cp_i*rcp_m*G[i][m]
      float tcol[CHK];
#pragma unroll
      for (int i = 0; i < CHK; ++i) {
        float acc = (i == lane) ? 1.0f : 0.0f;
        const float bc = sBet[i] * sCp[i];
#pragma unroll
        for (int m = 0; m < CHK; ++m) {
          if (m < i) {
            float Wim = bc * sRcp[m] * sG[i * TSTR + m];
            acc -= Wim * tcol[m];
          }
        }
        tcol[i] = acc;
        sT[i * TSTR + lane] = acc;
      }
    } else if (wave == 1) {
      for (int idx = lane; idx < CHK * CHK; idx += 32) {
        int i = idx >> 4, j = idx & 15;
        float val = 0.f;
        if (j <= i) val = sCp[i] * sRcp[j] * sHq[i * TSTR + j];
        sM[i * TSTR + j] = val;
      }
    }
    v8f c1 = {}, pp = {};
    {
      const int rbase = wave * 16;
#pragma unroll 8
      for (int kk = 0; kk < DKC; kk += 4) {
        v2f a;  a.x  = sS[(rbase + lid) * KSTR + kk + koff];
                a.y  = sS[(rbase + lid) * KSTR + kk + koff + 1];
        v2f bk; bk.x = sK[lid * KSTR + kk + koff]; bk.y = sK[lid * KSTR + kk + koff + 1];
        v2f bq; bq.x = sQ[lid * KSTR + kk + koff]; bq.y = sQ[lid * KSTR + kk + koff + 1];
        c1 = wmma4(a, bk, c1);
        pp = wmma4(a, bq, pp);
      }
    }
    __syncthreads();  // (4)

    // ---- phase C: U, Delta, O, state update (each wave: its own 16-row band)
    {
      const int rbase  = wave * 16;
      const float bt_i = sBet[lid];
      const float cp_i = sCp[lid];
      float* scr = sScr[wave];

      // U[r,i] = beta_i * (V[i][r] - cp_i * C1[r,i])  (accumulator layout)
#pragma unroll
      for (int m = 0; m < 8; ++m) {
        float u = bt_i * (sV[lid * VSTR + rbase + m + 8 * half] - cp_i * c1[m]);
        scr[(m + 8 * half) * TSTR + lid] = u;
      }
      // Delta = U * T^T   (K-dim = 16 tokens -> 4 WMMA steps)
      v8f dl = {};
#pragma unroll
      for (int kk = 0; kk < CHK; kk += 4) {
        v2f a;  a.x  = scr[lid * TSTR + kk + koff]; a.y  = scr[lid * TSTR + kk + koff + 1];
        v2f bb; bb.x = sT[lid * TSTR + kk + koff];  bb.y = sT[lid * TSTR + kk + koff + 1];
        dl = wmma4(a, bb, dl);
      }
#pragma unroll
      for (int m = 0; m < 8; ++m)
        scr[(m + 8 * half) * TSTR + lid] = dl[m];   // overwrite U with Delta

      // hoist Delta A-operand (and decay-scaled copy for state update)
      v2f aD[4], aDs[4];
#pragma unroll
      for (int t4 = 0; t4 < 4; ++t4) {
        int kk = t4 * 4 + koff;
        v2f a; a.x = scr[lid * TSTR + kk]; a.y = scr[lid * TSTR + kk + 1];
        aD[t4] = a;
        v2f as; as.x = a.x * sDs[kk]; as.y = a.y * sDs[kk + 1];
        aDs[t4] = as;
      }

      // O = diag(cp) P + Delta * M^T
      v8f oo;
#pragma unroll
      for (int m = 0; m < 8; ++m) oo[m] = cp_i * pp[m];
#pragma unroll
      for (int t4 = 0; t4 < 4; ++t4) {
        int kk = t4 * 4 + koff;
        v2f bb; bb.x = sM[lid * TSTR + kk]; bb.y = sM[lid * TSTR + kk + 1];
        oo = wmma4(aD[t4], bb, oo);
      }
      {
        float* op = oB + (size_t)(t0 + lid) * vTok + rbase + 8 * half;
        *(float4*)(op)     = make_float4(oo[0], oo[1], oo[2], oo[3]);
        *(float4*)(op + 4) = make_float4(oo[4], oo[5], oo[6], oo[7]);
      }

      // S_new = cp15 * S0 + Delta' * K   (8 N-tiles of 16 over DK)
      const float cpT = sCpT;
#pragma unroll 2
      for (int nt = 0; nt < DKC / 16; ++nt) {
        const int n0 = nt * 16;
        v8f s;
#pragma unroll
        for (int m = 0; m < 8; ++m)
          s[m] = cpT * sS[(rbase + m + 8 * half) * KSTR + n0 + lid];
#pragma unroll
        for (int t4 = 0; t4 < 4; ++t4) {
          int kk = t4 * 4 + koff;
          v2f bb; bb.x = sK[kk * KSTR + n0 + lid];
                  bb.y = sK[(kk + 1) * KSTR + n0 + lid];
          s = wmma4(aDs[t4], bb, s);
        }
#pragma unroll
        for (int m = 0; m < 8; ++m)
          sS[(rbase + m + 8 * half) * KSTR + n0 + lid] = s[m];
      }
    }
  }
}

extern "C" void kernel_launch(void* const* d_in, const int* in_sizes, int n_in,
                              void* d_out, int out_size, void* d_ws, size_t ws_size,
                              hipStream_t stream) {
  (void)in_sizes; (void)n_in; (void)out_size; (void)d_ws; (void)ws_size;
  const float* q    = (const float*)d_in[0];
  const float* k    = (const float*)d_in[1];
  const float* v    = (const float*)d_in[2];
  const float* g    = (const float*)d_in[3];
  const float* beta = (const float*)d_in[4];
  const float* st   = (const float*)d_in[5];
  float* out = (float*)d_out;
  dim3 grid(BB * HVC * (DVC / DVB));   // 128 workgroups
  dim3 block(NTHR);                    // 4 waves (wave32)
  gdn_chunked_kernel<<<grid, block, 0, stream>>>(q, k, v, g, beta, st, out);
}